// PCSRec_86079734546857
// MI455X (gfx1250) — compile-verified
//
#include <hip/hip_runtime.h>
#include <math.h>

typedef float v2f __attribute__((ext_vector_type(2)));
typedef float v8f __attribute__((ext_vector_type(8)));

#define NUM_USERS 80000
#define NUM_ITEMS 40000
#define NUM_NODES (NUM_USERS + NUM_ITEMS)
#define DIM 64
#define ALPHA_W 0.8f
#define NPATHS 6

// ---------------------------------------------------------------------------
// softmax over the 6-element theta vector -> theta_w
// ---------------------------------------------------------------------------
__global__ void pcs_softmax_theta(const float* __restrict__ theta,
                                  float* __restrict__ tw) {
  if (threadIdx.x == 0 && blockIdx.x == 0) {
    float m = theta[0];
    #pragma unroll
    for (int i = 1; i < NPATHS; ++i) m = fmaxf(m, theta[i]);
    float e0 = expf(theta[0] - m), e1 = expf(theta[1] - m), e2 = expf(theta[2] - m);
    float e3 = expf(theta[3] - m), e4 = expf(theta[4] - m), e5 = expf(theta[5] - m);
    float s = e0 + e1 + e2 + e3 + e4 + e5;
    tw[0] = e0 / s; tw[1] = e1 / s; tw[2] = e2 / s;
    tw[3] = e3 / s; tw[4] = e4 / s; tw[5] = e5 / s;
  }
}

// ---------------------------------------------------------------------------
// vals = p_counts @ theta_w  via V_WMMA_F32_16X16X4_F32 (fp32, K=6 as 4+2),
// then exp_v = exp(vals), row_sums[p_row] += exp_v  (atomic segment sum).
// One wave handles 16 edges. Layouts per CDNA5 ISA 7.12.2:
//   A (16x4 f32):  lane L, vgpr v -> A[L%16][v + 2*(L/16)]
//   B (4x16 f32):  lane L, vgpr v -> B[v + 2*(L/16)][L%16]  (theta replicated over N)
//   D (16x16 f32): lane L, vgpr v -> D[v + 8*(L/16)][L%16]
// ---------------------------------------------------------------------------
__global__ void pcs_pathvals_wmma(const float* __restrict__ pc,
                                  const int*   __restrict__ prow,
                                  const float* __restrict__ tw,
                                  float* __restrict__ ev,
                                  float* __restrict__ rowsums,
                                  int nE) {
  const int lane = threadIdx.x & 31;
  const int wid  = (int)(((unsigned)blockIdx.x * blockDim.x + threadIdx.x) >> 5);
  const int e0   = wid * 16;
  if (e0 >= nE) return;                       // wave-uniform
  const int half = lane >> 4;

  if (e0 + 16 <= nE) {                        // full tile: EXEC all ones
    const int m = lane & 15;
    const float* rp = pc + (size_t)(e0 + m) * NPATHS;

    v2f a1; a1.x = rp[2 * half];     a1.y = rp[2 * half + 1];   // K = 0..3
    v2f b1; b1.x = tw[2 * half];     b1.y = tw[2 * half + 1];
    v2f a2; a2.x = 0.f; a2.y = 0.f;                              // K = 4..7 (6,7 pad)
    v2f b2; b2.x = 0.f; b2.y = 0.f;
    if (half == 0) { a2.x = rp[4]; a2.y = rp[5]; b2.x = tw[4]; b2.y = tw[5]; }

    v8f c = {};
    c = __builtin_amdgcn_wmma_f32_16x16x4_f32(false, a1, false, b1, (short)0, c, false, false);
    c = __builtin_amdgcn_wmma_f32_16x16x4_f32(false, a2, false, b2, (short)0, c, false, false);

    // Branchless extraction of c[s], s = lane&7: balanced cndmask tree on bits.
    const int s  = lane & 7;
    const bool b0 = (s & 1) != 0;
    const bool b1s = (s & 2) != 0;
    const bool b2s = (s & 4) != 0;
    const float f0 = b0 ? c[1] : c[0];
    const float f1 = b0 ? c[3] : c[2];
    const float f2 = b0 ? c[5] : c[4];
    const float f3 = b0 ? c[7] : c[6];
    const float g0 = b1s ? f1 : f0;
    const float g1 = b1s ? f3 : f2;
    const float v  = b2s ? g1 : g0;

    // edge (s + 8*half): handled once by lanes 0-7 and 16-23
    if ((lane & 15) < 8) {
      const int edge = e0 + s + 8 * half;
      const float x = expf(v);
      ev[edge] = x;
      atomicAdd(rowsums + prow[edge], x);
    }
  } else {                                    // ragged tail: scalar per lane
    const int edge = e0 + lane;
    if (edge < nE) {
      const float* rp = pc + (size_t)edge * NPATHS;
      float v = 0.f;
      #pragma unroll
      for (int k = 0; k < NPATHS; ++k) v = fmaf(rp[k], tw[k], v);
      const float x = expf(v);
      ev[edge] = x;
      atomicAdd(rowsums + prow[edge], x);
    }
  }
}

// soft_v = exp_v / (row_sums[p_row] + 1e-12), in place
__global__ void pcs_softv(float* __restrict__ ev, const float* __restrict__ rowsums,
                          const int* __restrict__ prow, int nE) {
  const int i = blockIdx.x * blockDim.x + threadIdx.x;
  if (i < nE) ev[i] = ev[i] / (rowsums[prow[i]] + 1e-12f);
}

// ---------------------------------------------------------------------------
// SpMM with register-accumulated segment sum. Edge lists are sorted by row
// (np.unique), so each wave accumulates its 64-wide row in registers and only
// touches memory with atomics on row changes / chunk boundaries.
// Lane l owns dims [2l, 2l+1]. Gather source: col<split -> xu, else xi.
// ---------------------------------------------------------------------------
__global__ void pcs_spmm(const int* __restrict__ row, const int* __restrict__ col,
                         const float* __restrict__ val,
                         const float* __restrict__ xu, const float* __restrict__ xi,
                         int split, float* __restrict__ y, int nE, int wavesTotal) {
  const int lane = threadIdx.x & 31;
  const long wid = (long)(((unsigned)blockIdx.x * blockDim.x + threadIdx.x) >> 5);
  const int per = (nE + wavesTotal - 1) / wavesTotal;
  long e = wid * (long)per;
  if (e >= nE) return;
  long eEnd = e + per; if (eEnd > nE) eEnd = nE;

  float ax = 0.f, ay = 0.f;
  int cur = __builtin_amdgcn_readfirstlane(row[e]);
  for (; e < eEnd; ++e) {
    const int r = __builtin_amdgcn_readfirstlane(row[e]);
    if (r != cur) {
      atomicAdd(y + (size_t)cur * DIM + 2 * lane,     ax);
      atomicAdd(y + (size_t)cur * DIM + 2 * lane + 1, ay);
      ax = 0.f; ay = 0.f; cur = r;
    }
    const int ci = __builtin_amdgcn_readfirstlane(col[e]);
    const float v = val[e];
    const float* xr = (ci < split) ? (xu + (size_t)ci * DIM)
                                   : (xi + (size_t)(ci - split) * DIM);
    const float2 xv = *(const float2*)(xr + 2 * lane);
    ax = fmaf(v, xv.x, ax);
    ay = fmaf(v, xv.y, ay);
  }
  atomicAdd(y + (size_t)cur * DIM + 2 * lane,     ax);
  atomicAdd(y + (size_t)cur * DIM + 2 * lane + 1, ay);
}

// acc = e  (initialize mean accumulator with e0)
__global__ void pcs_copy4(const float4* __restrict__ src, float4* __restrict__ dst, int n4) {
  const int i = blockIdx.x * blockDim.x + threadIdx.x;
  if (i < n4) dst[i] = src[i];
}

// t = e_pos + alpha*(e - e_neg); e = t; acc += t
__global__ void pcs_combine4(const float4* __restrict__ ep, const float4* __restrict__ en,
                             float4* __restrict__ e, float4* __restrict__ acc, int n4) {
  const int i = blockIdx.x * blockDim.x + threadIdx.x;
  if (i < n4) {
    const float4 p = ep[i], g = en[i], cu = e[i];
    float4 a = acc[i], t;
    t.x = p.x + ALPHA_W * (cu.x - g.x);
    t.y = p.y + ALPHA_W * (cu.y - g.y);
    t.z = p.z + ALPHA_W * (cu.z - g.z);
    t.w = p.w + ALPHA_W * (cu.w - g.w);
    e[i] = t;
    a.x += t.x; a.y += t.y; a.z += t.z; a.w += t.w;
    acc[i] = a;
  }
}

__global__ void pcs_scale4(float4* __restrict__ acc, float s, int n4) {
  const int i = blockIdx.x * blockDim.x + threadIdx.x;
  if (i < n4) {
    float4 a = acc[i];
    a.x *= s; a.y *= s; a.z *= s; a.w *= s;
    acc[i] = a;
  }
}

// ---------------------------------------------------------------------------
extern "C" void kernel_launch(void* const* d_in, const int* in_sizes, int n_in,
                              void* d_out, int out_size, void* d_ws, size_t ws_size,
                              hipStream_t stream) {
  const float* user_emb = (const float*)d_in[0];
  const float* item_emb = (const float*)d_in[1];
  const float* theta    = (const float*)d_in[2];
  const int*   pos_row  = (const int*)  d_in[3];
  const int*   pos_col  = (const int*)  d_in[4];
  const float* pos_val  = (const float*)d_in[5];
  const int*   neg_row  = (const int*)  d_in[6];
  const int*   neg_col  = (const int*)  d_in[7];
  const float* neg_val  = (const float*)d_in[8];
  const int*   p_row    = (const int*)  d_in[9];
  const int*   p_col    = (const int*)  d_in[10];
  const float* p_counts = (const float*)d_in[11];

  const int ePos  = in_sizes[3];
  const int eNeg  = in_sizes[6];
  const int ePath = in_sizes[9];
  const size_t nfeat = (size_t)NUM_NODES * DIM;

  // workspace carve-out (256B aligned slices)
  char* w = (char*)d_ws;
  auto carve = [&](size_t bytes) -> float* {
    float* p = (float*)w;
    w += (bytes + 255) & ~(size_t)255;
    return p;
  };
  float* tw      = carve(8 * sizeof(float));
  float* rowsums = carve((size_t)NUM_NODES * sizeof(float));
  float* sv      = carve((size_t)ePath * sizeof(float));   // exp_v then soft_v
  float* e       = carve(nfeat * sizeof(float));
  float* epos    = carve(nfeat * sizeof(float));
  float* eneg    = carve(nfeat * sizeof(float));
  float* acc     = (float*)d_out;

  const int n4 = (int)(nfeat / 4);
  const int ELT_BLK = 256;
  const int elt_grid = (n4 + ELT_BLK - 1) / ELT_BLK;

  // 1. theta softmax + zero row sums
  hipMemsetAsync(rowsums, 0, (size_t)NUM_NODES * sizeof(float), stream);
  pcs_softmax_theta<<<1, 32, 0, stream>>>(theta, tw);

  // 2. path values via f32 WMMA, exp, atomic row sums
  const int wavesP = (ePath + 15) / 16;
  const int thrP   = wavesP * 32;
  pcs_pathvals_wmma<<<(thrP + 255) / 256, 256, 0, stream>>>(p_counts, p_row, tw, sv,
                                                            rowsums, ePath);

  // 3. edge softmax normalization (in place)
  pcs_softv<<<(ePath + 255) / 256, 256, 0, stream>>>(sv, rowsums, p_row, ePath);

  // 4. e0 = spmm(p, soft_v, concat(user_emb, item_emb))
  const int SPMM_BLOCKS = 4096, SPMM_THREADS = 256;
  const int wavesTotal = SPMM_BLOCKS * SPMM_THREADS / 32;
  hipMemsetAsync(e, 0, nfeat * sizeof(float), stream);
  pcs_spmm<<<SPMM_BLOCKS, SPMM_THREADS, 0, stream>>>(p_row, p_col, sv, user_emb, item_emb,
                                                     NUM_USERS, e, ePath, wavesTotal);

  // 5. acc = e0
  pcs_copy4<<<elt_grid, ELT_BLK, 0, stream>>>((const float4*)e, (float4*)acc, n4);

  // 6. three propagation layers
  for (int layer = 0; layer < 3; ++layer) {
    hipMemsetAsync(epos, 0, nfeat * sizeof(float), stream);
    hipMemsetAsync(eneg, 0, nfeat * sizeof(float), stream);
    pcs_spmm<<<SPMM_BLOCKS, SPMM_THREADS, 0, stream>>>(pos_row, pos_col, pos_val, e, e,
                                                       NUM_NODES + 1, epos, ePos, wavesTotal);
    pcs_spmm<<<SPMM_BLOCKS, SPMM_THREADS, 0, stream>>>(neg_row, neg_col, neg_val, e, e,
                                                       NUM_NODES + 1, eneg, eNeg, wavesTotal);
    pcs_combine4<<<elt_grid, ELT_BLK, 0, stream>>>((const float4*)epos, (const float4*)eneg,
                                                   (float4*)e, (float4*)acc, n4);
  }

  // 7. mean over the 4 snapshots
  pcs_scale4<<<elt_grid, ELT_BLK, 0, stream>>>((float4*)acc, 0.25f, n4);
}